// ObservationEncoder_30674656428108
// MI455X (gfx1250) — compile-verified
//
#include <hip/hip_runtime.h>
#include <hip/hip_bf16.h>

typedef float v2f __attribute__((ext_vector_type(2)));
typedef float v8f __attribute__((ext_vector_type(8)));
typedef unsigned int u32x4 __attribute__((ext_vector_type(4)));
typedef int i32x4 __attribute__((ext_vector_type(4)));
typedef int i32x8 __attribute__((ext_vector_type(8)));

#define DIM    1024
#define NB     1024
#define KCODES 512
#define EMB    256
#define NTOK   4096   // (NB*DIM)/EMB

#define LDSPAD 36     // padded row stride (dwords) for both LDS tiles
#define KCHUNK 32     // K staged per iteration
#define NCHUNK (DIM / KCHUNK)

#if defined(__has_builtin)
#  if __has_builtin(__builtin_amdgcn_tensor_load_to_lds) && __has_builtin(__builtin_amdgcn_s_wait_tensorcnt)
#    define HAVE_TDM 1
#  else
#    define HAVE_TDM 0
#  endif
#else
#  define HAVE_TDM 0
#endif

#if HAVE_TDM
// ---------------------------------------------------------------------------
// TDM: DMA a 2D f32 tile (tileD0 x tileD1, row stride strideElems) from global
// memory into LDS at ldsByteAddr, inserting 4 pad dwords after every 32 dwords
// (=> LDS row stride of 36 dwords). Descriptor per CDNA5 ISA 8.3/8.4.
// ---------------------------------------------------------------------------
__device__ __forceinline__ void tdm_load_tile_f32(const float* gbase,
                                                  unsigned ldsByteAddr,
                                                  unsigned tileD0, unsigned tileD1,
                                                  unsigned tensorD0, unsigned tensorD1,
                                                  unsigned strideElems)
{
    const unsigned long long ga = (unsigned long long)(uintptr_t)gbase;
    u32x4 g0;
    g0[0] = 1u;                                     // count=1 (valid), user mode
    g0[1] = ldsByteAddr;                            // lds_addr
    g0[2] = (unsigned)(ga & 0xFFFFFFFFu);           // global_addr[31:0]
    g0[3] = (unsigned)((ga >> 32) & 0x01FFFFFFu)    // global_addr[56:32]
          | (2u << 30);                             // type=2 ("image")
    i32x8 g1;
    g1[0] = (int)((2u << 16)        // data_size = 4B
                | (1u << 20)        // pad_enable
                | (4u << 22)        // pad_interval: every 32 dwords
                | (3u << 25));      // pad_amount: 4 dwords  -> stride 36 dwords
    g1[1] = (int)((tensorD0 & 0xFFFFu) << 16);                    // dim0[15:0]
    g1[2] = (int)((tensorD0 >> 16) | ((tensorD1 & 0xFFFFu) << 16));
    g1[3] = (int)((tensorD1 >> 16) | (tileD0 << 16));             // tile_dim0
    g1[4] = (int)(tileD1 & 0xFFFFu);                              // tile_dim1 (tile_dim2=0)
    g1[5] = (int)strideElems;                                     // dim0_stride[31:0]
    g1[6] = 0;
    g1[7] = 0;
    const i32x4 z4 = {0, 0, 0, 0};
#if defined(__clang_major__) && (__clang_major__ >= 23)
    const i32x8 z8 = {0, 0, 0, 0, 0, 0, 0, 0};
    __builtin_amdgcn_tensor_load_to_lds(g0, g1, z4, z4, z8, 0);
#else
    __builtin_amdgcn_tensor_load_to_lds(g0, g1, z4, z4, 0);
#endif
}
#endif // HAVE_TDM

// ---------------------------------------------------------------------------
// GEMM: C[1024x1024] = act(A[1024x1024] @ W[1024x1024] + bias)
// Block tile 128(M) x 64(N); K in 32-wide chunks, double-buffered in LDS.
// 256 threads = 8 waves; wave w computes rows [w*16,w*16+16) x all 64 cols
// (4 accumulators) via v_wmma_f32_16x16x4_f32.
// Pipeline per chunk c: issue TDM(A,c+1) + global loads(B,c+1) -> compute(c)
// under the in-flight traffic -> drain B regs to LDS -> wait tensorcnt ->
// single barrier -> flip buffers.
// ---------------------------------------------------------------------------
template <bool RELU>
__global__ __launch_bounds__(256)
void gemm_wmma_f32(const float* __restrict__ A, const float* __restrict__ W,
                   const float* __restrict__ bias, float* __restrict__ C)
{
    __shared__ float As[2][128 * LDSPAD];   // A tile [row][k], padded, x2 buffers
    __shared__ float BsT[2][64 * LDSPAD];   // B tile TRANSPOSED [n][k], padded, x2

    const int tid  = threadIdx.x;
    const int lane = tid & 31;
    const int wave = tid >> 5;
    const int hi   = lane >> 4;             // 0 = lanes 0-15, 1 = lanes 16-31
    const int l16  = lane & 15;

    const int mBase = blockIdx.y * 128;
    const int nBase = blockIdx.x * 64;
    const int mWave = wave * 16;

    // This thread's B-staging coordinates (2 float4 per thread per chunk)
    const int lin0 = tid * 4;               // t = 0
    const int lin1 = (tid + 256) * 4;       // t = 1
    const int br0 = lin0 >> 6, bc0 = lin0 & 63;
    const int br1 = lin1 >> 6, bc1 = lin1 & 63;

    v8f acc[4];
#pragma unroll
    for (int t = 0; t < 4; ++t)
#pragma unroll
        for (int e = 0; e < 8; ++e) acc[t][e] = 0.0f;

    // ---- helpers -----------------------------------------------------------
    auto stageA = [&](int k0, int buf) {
#if HAVE_TDM
        // Each wave DMAs its own 16-row slice (one TDM op per wave).
        tdm_load_tile_f32(&A[(size_t)(mBase + mWave) * DIM + k0],
                          (unsigned)(uintptr_t)&As[buf][0]
                              + (unsigned)(mWave * LDSPAD * 4),
                          /*tileD0=*/KCHUNK, /*tileD1=*/16,
                          /*tensorD0=*/DIM, /*tensorD1=*/DIM,
                          /*strideElems=*/DIM);
#else
#pragma unroll
        for (int t = 0; t < 4; ++t) {
            const int linear = (tid + t * 256) * 4;
            const int r = linear >> 5;
            const int c = linear & 31;
            const float4 v = *reinterpret_cast<const float4*>(
                &A[(size_t)(mBase + r) * DIM + k0 + c]);
            As[buf][r * LDSPAD + c + 0] = v.x; As[buf][r * LDSPAD + c + 1] = v.y;
            As[buf][r * LDSPAD + c + 2] = v.z; As[buf][r * LDSPAD + c + 3] = v.w;
        }
#endif
    };
    auto loadB = [&](int k0, float4* breg) {
        breg[0] = *reinterpret_cast<const float4*>(
            &W[(size_t)(k0 + br0) * DIM + nBase + bc0]);
        breg[1] = *reinterpret_cast<const float4*>(
            &W[(size_t)(k0 + br1) * DIM + nBase + bc1]);
    };
    auto storeB = [&](const float4* breg, int buf) {
        float* bs = &BsT[buf][0];
        bs[(bc0 + 0) * LDSPAD + br0] = breg[0].x;
        bs[(bc0 + 1) * LDSPAD + br0] = breg[0].y;
        bs[(bc0 + 2) * LDSPAD + br0] = breg[0].z;
        bs[(bc0 + 3) * LDSPAD + br0] = breg[0].w;
        bs[(bc1 + 0) * LDSPAD + br1] = breg[1].x;
        bs[(bc1 + 1) * LDSPAD + br1] = breg[1].y;
        bs[(bc1 + 2) * LDSPAD + br1] = breg[1].z;
        bs[(bc1 + 3) * LDSPAD + br1] = breg[1].w;
    };

    // ---- prologue: stage chunk 0 into buffer 0 -----------------------------
    {
        float4 breg[2];
        stageA(0, 0);
        loadB(0, breg);
        storeB(breg, 0);
#if HAVE_TDM
        __builtin_amdgcn_s_wait_tensorcnt(0);
#endif
        __syncthreads();
    }

    // ---- main K loop -------------------------------------------------------
    for (int c = 0; c < NCHUNK; ++c) {
        const int cur = c & 1;
        float4 breg[2];
        const bool more = (c + 1 < NCHUNK);
        if (more) {
            stageA((c + 1) * KCHUNK, cur ^ 1);   // async DMA under compute
            loadB((c + 1) * KCHUNK, breg);       // global loads in flight
        }

        // ---- 32 WMMAs on buffer `cur`, fragment-pipelined ----
        const float* as = &As[cur][0];
        const float* bs = &BsT[cur][0];
        const int aRow = (mWave + l16) * LDSPAD + 2 * hi;

        v2f a0, b0[4], a1, b1[4];
        a0 = *reinterpret_cast<const v2f*>(&as[aRow]);
#pragma unroll
        for (int nt = 0; nt < 4; ++nt)
            b0[nt] = *reinterpret_cast<const v2f*>(
                &bs[(nt * 16 + l16) * LDSPAD + 2 * hi]);

#pragma unroll
        for (int kk = 0; kk < KCHUNK; kk += 4) {
            if (kk + 4 < KCHUNK) {  // prefetch next step's fragments
                a1 = *reinterpret_cast<const v2f*>(&as[aRow + kk + 4]);
#pragma unroll
                for (int nt = 0; nt < 4; ++nt)
                    b1[nt] = *reinterpret_cast<const v2f*>(
                        &bs[(nt * 16 + l16) * LDSPAD + 2 * hi + kk + 4]);
            }
#pragma unroll
            for (int nt = 0; nt < 4; ++nt)
                acc[nt] = __builtin_amdgcn_wmma_f32_16x16x4_f32(
                    false, a0, false, b0[nt], (short)0, acc[nt], false, false);
            if (kk + 4 < KCHUNK) {
                a0 = a1;
#pragma unroll
                for (int nt = 0; nt < 4; ++nt) b0[nt] = b1[nt];
            }
        }

        if (more) {
            storeB(breg, cur ^ 1);               // drain staged B into LDS
#if HAVE_TDM
            __builtin_amdgcn_s_wait_tensorcnt(0);
#endif
            __syncthreads();                     // single barrier per chunk
        }
    }

    // ---- epilogue: bias (+ReLU), scatter per C/D layout --------------------
#pragma unroll
    for (int nt = 0; nt < 4; ++nt) {
        const int col  = nBase + nt * 16 + l16;
        const float bv = bias[col];
#pragma unroll
        for (int r = 0; r < 8; ++r) {
            const int row = mBase + mWave + hi * 8 + r;
            float v = acc[nt][r] + bv;
            if (RELU) v = fmaxf(v, 0.0f);
            C[(size_t)row * DIM + col] = v;
        }
    }
}

// ---------------------------------------------------------------------------
// Codebook squared norms (512 codes x 256 dims) + zero the loss accumulator.
// ---------------------------------------------------------------------------
__global__ __launch_bounds__(256)
void cnorm_kernel(const float* __restrict__ cb, float* __restrict__ cnorm,
                  float* __restrict__ loss)
{
    const int code = blockIdx.x * 256 + threadIdx.x;
    if (blockIdx.x == 0 && threadIdx.x == 0) *loss = 0.0f;
    if (code < KCODES) {
        const float4* row = reinterpret_cast<const float4*>(cb + (size_t)code * EMB);
        float s = 0.0f;
#pragma unroll 8
        for (int d4 = 0; d4 < EMB / 4; ++d4) {
            const float4 c = row[d4];
            s += c.x * c.x + c.y * c.y + c.z * c.z + c.w * c.w;
        }
        cnorm[code] = s;
    }
}

// ---------------------------------------------------------------------------
// VQ: one wave per token. argmin_k (||c_k||^2 - 2 z.c_k), gather winner,
// accumulate 1.25 * mean((q-z)^2), emit index.
// ---------------------------------------------------------------------------
__global__ __launch_bounds__(256)
void vq_kernel(const float* __restrict__ feat, const float* __restrict__ cb,
               const float* __restrict__ cnorm, float* __restrict__ out,
               float* __restrict__ loss, float* __restrict__ idx_out)
{
    __shared__ float zs[8][EMB];

    const int tid   = threadIdx.x;
    const int lane  = tid & 31;
    const int wave  = tid >> 5;
    const int token = blockIdx.x * 8 + wave;

    const float* z = feat + (size_t)token * EMB;
#pragma unroll
    for (int t = 0; t < 8; ++t)
        zs[wave][lane + t * 32] = z[lane + t * 32];
    // wave-private LDS slice: same-wave DS ordering suffices

    float bestD = 3.4e38f;
    int   bestI = 0;
    const float4* zrow = reinterpret_cast<const float4*>(zs[wave]);
    for (int j = 0; j < KCODES / 32; ++j) {
        const int code = lane + j * 32;
        const float4* crow = reinterpret_cast<const float4*>(cb + (size_t)code * EMB);
        float dot = 0.0f;
#pragma unroll 8
        for (int d4 = 0; d4 < EMB / 4; ++d4) {
            const float4 c = crow[d4];
            const float4 v = zrow[d4];
            dot += c.x * v.x + c.y * v.y + c.z * v.z + c.w * v.w;
        }
        const float dist = cnorm[code] - 2.0f * dot;
        if (dist < bestD || (dist == bestD && code < bestI)) {
            bestD = dist; bestI = code;
        }
    }
    // wave-wide argmin with lowest-index tie-break
#pragma unroll
    for (int off = 16; off > 0; off >>= 1) {
        const float od = __shfl_down(bestD, off, 32);
        const int   oi = __shfl_down(bestI, off, 32);
        if (od < bestD || (od == bestD && oi < bestI)) { bestD = od; bestI = oi; }
    }
    bestI = __shfl(bestI, 0, 32);

    const float* q = cb + (size_t)bestI * EMB;
    float lsum = 0.0f;
#pragma unroll
    for (int t = 0; t < 8; ++t) {
        const int d  = lane + t * 32;
        const float qv = q[d];
        const float zv = zs[wave][d];
        out[(size_t)token * EMB + d] = qv;
        const float diff = qv - zv;
        lsum += diff * diff;
    }
#pragma unroll
    for (int off = 16; off > 0; off >>= 1)
        lsum += __shfl_down(lsum, off, 32);

    if (lane == 0) {
        atomicAdd(loss, lsum * (1.25f / (float)(NTOK * EMB)));
        idx_out[token] = (float)bestI;
    }
}

// ---------------------------------------------------------------------------
extern "C" void kernel_launch(void* const* d_in, const int* in_sizes, int n_in,
                              void* d_out, int out_size, void* d_ws, size_t ws_size,
                              hipStream_t stream)
{
    const float* obs = (const float*)d_in[0];
    const float* W1  = (const float*)d_in[1];
    const float* b1  = (const float*)d_in[2];
    const float* W2  = (const float*)d_in[3];
    const float* b2  = (const float*)d_in[4];
    const float* W3  = (const float*)d_in[5];
    const float* b3  = (const float*)d_in[6];
    const float* cb  = (const float*)d_in[7];

    float* out  = (float*)d_out;                     // [1024*1024]
    float* loss = out + (size_t)NB * DIM;            // [1]
    float* idxf = loss + 1;                          // [4096]

    float* h1    = (float*)d_ws;                     // 1M floats
    float* h2    = h1 + (size_t)NB * DIM;            // 1M floats
    float* feat  = h2 + (size_t)NB * DIM;            // 1M floats
    float* cnorm = feat + (size_t)NB * DIM;          // 512 floats

    const dim3 grid(DIM / 64, NB / 128);             // N-tiles x M-tiles
    const dim3 block(256);

    gemm_wmma_f32<true ><<<grid, block, 0, stream>>>(obs, W1, b1, h1);
    gemm_wmma_f32<true ><<<grid, block, 0, stream>>>(h1,  W2, b2, h2);
    gemm_wmma_f32<false><<<grid, block, 0, stream>>>(h2,  W3, b3, feat);

    cnorm_kernel<<<2, 256, 0, stream>>>(cb, cnorm, loss);
    vq_kernel<<<NTOK / 8, 256, 0, stream>>>(feat, cb, cnorm, out, loss, idxf);
}